// EmbeddingFreeContextAttention_72507637891614
// MI455X (gfx1250) — compile-verified
//
#include <hip/hip_runtime.h>

// ---------------------------------------------------------------------------
// EmbeddingFreeContextAttention for MI455X (gfx1250, wave32)
//   out[b,l,0:D]   = sent[b,l,:] * <sent[b,idx0,:], sent[b,l,:]>
//   out[b,l,D:2D]  = sent[b,l,:] * <sent[b,idx1,:], sent[b,l,:]>
// Memory-bound (~151 MB @ 23.3 TB/s ~= 6.5 us). Scores computed on the
// matrix pipe with V_WMMA_F32_16X16X4_F32 (full f32 = reference precision).
// Round-2 changes: constant-trip scalar-controlled WMMA loop (no EXEC
// masking), unroll-by-4, non-temporal streaming loads/stores for the
// single-use tile and write-once output.
// ---------------------------------------------------------------------------

typedef __attribute__((ext_vector_type(2))) float v2f;
typedef __attribute__((ext_vector_type(4))) float v4f;
typedef __attribute__((ext_vector_type(8))) float v8f;

#define B_DIM 32
#define L_DIM 512
#define D_DIM 768
#define TILE 16
#define KSTEPS (D_DIM / 4)       // 192 WMMA k-steps of K=4 (96 per wave)
#define QROW (D_DIM / 4)         // 192 float4s per token row
#define ROWSTRIDE 772            // 768 + 4 pad floats: rows land on distinct
                                 // LDS banks (772%64==4) and rows stay 16B
                                 // aligned (772*4 = 3088 = 193*16)

__global__ __launch_bounds__(64) void efca_kernel(
    const float* __restrict__ sent,   // [B, L, D]
    const int*   __restrict__ idxs,   // [B, 2]
    float*       __restrict__ out) {  // [B, L, 2D]

  __shared__ float lds_tile[TILE * ROWSTRIDE];   // 16 x 768 token tile
  __shared__ float lds_e[2 * D_DIM];             // two entity vectors
  __shared__ float lds_attp[2][2][TILE];         // per-wave partial scores
  __shared__ float lds_att[2][TILE];             // combined scores

  const int tid  = threadIdx.x;
  const int lane = tid & 31;
  // Force the wave id into an SGPR so the WMMA loop is scalar-controlled.
  const int wave = __builtin_amdgcn_readfirstlane(tid) >> 5;   // 0..1
  const int b    = blockIdx.x >> 5;   // gridDim.x = 32 * 32
  const int tile = blockIdx.x & 31;
  const int l0   = tile * TILE;

  const float* tile_base = sent + (size_t)(b * L_DIM + l0) * D_DIM;

  // ---- stage entity vectors (coalesced; keep in cache: reused 32x/batch) -
  const int i0 = idxs[b * 2 + 0];
  const int i1 = idxs[b * 2 + 1];
  const float* e0p = sent + (size_t)(b * L_DIM + i0) * D_DIM;
  const float* e1p = sent + (size_t)(b * L_DIM + i1) * D_DIM;
  for (int d = tid; d < D_DIM; d += 64) {
    lds_e[d]         = e0p[d];
    lds_e[D_DIM + d] = e1p[d];
  }

  // ---- stage 16x768 tile: coalesced 128-bit non-temporal loads -----------
  for (int i = tid; i < TILE * QROW; i += 64) {
    const int m = i / QROW;
    const int q = i - m * QROW;
    const v4f v = __builtin_nontemporal_load(
        (const v4f*)(tile_base + (size_t)m * D_DIM) + q);
    *((v4f*)(lds_tile + m * ROWSTRIDE + q * 4)) = v;
  }
  __syncthreads();

  // ---- phase 1: att = tokens(16xD) x entities(Dx16) via f32 WMMA ---------
  // A 16x4 layout: lane 0-15 -> M=lane, VGPR j holds K = j + 2*(lane/16)
  // B 4x16 layout: lane 0-15 -> N=lane, VGPR j holds K = j + 2*(lane/16)
  // Entity columns >= 2 are value-masked to zero (no EXEC divergence:
  // WMMA requires EXEC all-ones).
  const int   row   = lane & 15;      // M for A, N for B
  const int   half  = lane >> 4;      // selects which K pair this lane holds
  const int   nsel  = (row < 2) ? row : 0;
  const float nmask = (row < 2) ? 1.0f : 0.0f;

  v8f C = {};
#pragma unroll 4
  for (int kk = 0; kk < KSTEPS / 2; ++kk) {      // constant trip count: 96
    const int ks  = (kk << 1) | wave;            // this wave's k-step
    const int col = ks * 4 + half * 2;
    v2f A;
    A.x = lds_tile[row * ROWSTRIDE + col];
    A.y = lds_tile[row * ROWSTRIDE + col + 1];
    v2f Bm;
    Bm.x = lds_e[nsel * D_DIM + col]     * nmask;
    Bm.y = lds_e[nsel * D_DIM + col + 1] * nmask;
    C = __builtin_amdgcn_wmma_f32_16x16x4_f32(
        /*neg_a=*/false, A, /*neg_b=*/false, Bm,
        /*c_mod=*/(short)0, C, /*reuse_a=*/false, /*reuse_b=*/false);
  }

  // C layout: VGPR r, lanes 0-15 -> M=r, lanes 16-31 -> M=r+8; N = lane&15.
  if (row < 2) {
#pragma unroll
    for (int r = 0; r < 8; ++r) {
      lds_attp[wave][row][r + 8 * half] = C[r];
    }
  }
  __syncthreads();

  if (tid < 32) {
    const int n = tid >> 4, m = tid & 15;
    lds_att[n][m] = lds_attp[0][n][m] + lds_attp[1][n][m];
  }
  __syncthreads();

  // ---- phase 2: scale tile by per-token scalars; NT streaming stores -----
  for (int i = tid; i < TILE * QROW; i += 64) {
    const int m = i / QROW;
    const int q = i - m * QROW;
    const v4f  s  = *((const v4f*)(lds_tile + m * ROWSTRIDE + q * 4));
    const float a0 = lds_att[0][m];
    const float a1 = lds_att[1][m];
    float* obase = out + (size_t)(b * L_DIM + l0 + m) * (2 * D_DIM);
    __builtin_nontemporal_store(s * a0, (v4f*)obase + q);
    __builtin_nontemporal_store(s * a1, (v4f*)(obase + D_DIM) + q);
  }
}

extern "C" void kernel_launch(void* const* d_in, const int* in_sizes, int n_in,
                              void* d_out, int out_size, void* d_ws, size_t ws_size,
                              hipStream_t stream) {
  const float* sent = (const float*)d_in[0];
  const int*   idxs = (const int*)d_in[1];
  float*       out  = (float*)d_out;
  (void)in_sizes; (void)n_in; (void)out_size; (void)d_ws; (void)ws_size;

  const int grid = B_DIM * (L_DIM / TILE);  // 1024 blocks, 64 threads each
  efca_kernel<<<grid, 64, 0, stream>>>(sent, idxs, out);
}